// SnakeSimpleLoss_8546984919404
// MI455X (gfx1250) — compile-verified
//
#include <hip/hip_runtime.h>
#include <math.h>

// ---------------------------------------------------------------------------
// SnakeSimpleLoss for MI455X (gfx1250, wave32, WMMA)
// Pipeline: build L -> LL=L@L (WMMA) -> assemble [I+tA | I] -> Gauss-Jordan
//           -> 9x9 gradient conv (pred, |pred|) -> 50-step snake evolution
//           (bilinear gathers + WMMA matmuls with register-resident A-tiles)
//           -> render min-distance map + deterministic MSE reduction.
// ---------------------------------------------------------------------------

#define HH 384
#define WW 384
#define HWSZ (HH * WW)
#define BB 4
#define NN 128
#define NSTEPS_ 50

typedef __attribute__((ext_vector_type(2))) float v2f;
typedef __attribute__((ext_vector_type(8))) float v8f;

// ----- workspace layout (floats) -------------------------------------------
constexpr int OFF_L     = 0;          // 128*128
constexpr int OFF_LL    = 16384;      // 128*128
constexpr int OFF_AUG   = 32768;      // AUG1 (128x256) then AUG2 (128x256)
constexpr int OFF_NOUT  = 98304;      // B*128*3
constexpr int OFF_PART  = 99840;      // 2304 partial sums (pad to 4096)
constexpr int OFF_GIMG  = 103936;     // B*4*H*W

// ---------------------------------------------------------------------------
// 1) L (tridiagonal snake Laplacian)
// ---------------------------------------------------------------------------
__global__ void build_L_kernel(float* __restrict__ Lm) {
    int idx = blockIdx.x * 256 + threadIdx.x;       // 16384 entries
    int i = idx >> 7, j = idx & 127;
    float v = 0.0f;
    if (i == j)                 v = (i == 0 || i == 127) ? 1.0f : 2.0f;
    else if (i - j == 1 || j - i == 1) v = -1.0f;
    Lm[idx] = v;
}

// ---------------------------------------------------------------------------
// 2) LL = L @ L   — V_WMMA_F32_16X16X4_F32, one wave per 16x16 output tile.
//    A-tile (16x4 f32): lane L holds A[L%16][ (L/16)*2 + {0,1} ]
//    B-tile (4x16 f32): lane L holds B[ (L/16)*2 + {0,1} ][L%16]
//    D (16x16 f32):     vgpr v, lane L -> D[v + (L>=16?8:0)][L%16]
// ---------------------------------------------------------------------------
__global__ void ll_matmul_kernel(const float* __restrict__ Lm, float* __restrict__ LL) {
    const int lane  = threadIdx.x & 31;
    const int tileR = blockIdx.x >> 3;
    const int tileC = blockIdx.x & 7;
    const int m     = lane & 15;
    const int r     = tileR * 16 + m;
    const int n     = tileC * 16 + m;
    const int khalf = (lane >> 4) * 2;
    v8f acc = {0.f, 0.f, 0.f, 0.f, 0.f, 0.f, 0.f, 0.f};
    for (int kb = 0; kb < 32; ++kb) {
        const int k = kb * 4 + khalf;
        v2f a, b;
        a.x = Lm[r * 128 + k];
        a.y = Lm[r * 128 + k + 1];
        b.x = Lm[k * 128 + n];
        b.y = Lm[(k + 1) * 128 + n];
        acc = __builtin_amdgcn_wmma_f32_16x16x4_f32(false, a, false, b,
                                                    (short)0, acc, false, false);
    }
    const int rbase = tileR * 16 + ((lane >> 4) << 3);
#pragma unroll
    for (int v = 0; v < 8; ++v)
        LL[(rbase + v) * 128 + n] = acc[v];
}

// ---------------------------------------------------------------------------
// 3) Assemble augmented systems [S | I]:
//    AUG1: S = I + 0.1*(0.01*L + 0.01*LL)   (for M)
//    AUG2: S = I + 0.1*0.01*L               (for Mw)
// ---------------------------------------------------------------------------
__global__ void assemble_kernel(const float* __restrict__ Lm,
                                const float* __restrict__ LL,
                                float* __restrict__ aug) {
    int idx = blockIdx.x * 256 + threadIdx.x;       // 65536 entries
    int mat = idx >> 15;
    int e   = idx & 32767;
    int row = e >> 8;
    int c   = e & 255;
    float v;
    if (c < 128) {
        float base = (row == c) ? 1.0f : 0.0f;
        float l    = Lm[row * 128 + c];
        if (mat == 0) v = base + 0.1f * (0.01f * l + 0.01f * LL[row * 128 + c]);
        else          v = base + 0.1f * 0.01f * l;
    } else {
        v = ((c - 128) == row) ? 1.0f : 0.0f;
    }
    aug[idx] = v;
}

// ---------------------------------------------------------------------------
// 4) Gauss-Jordan inversion (no pivoting; S is diagonally dominant).
//    One block per matrix; thread t owns column t of the 128x256 augmented.
// ---------------------------------------------------------------------------
__global__ void gauss_jordan_kernel(float* __restrict__ augBase) {
    float* A = augBase + blockIdx.x * 32768;
    __shared__ float mcol[128];
    const int t = threadIdx.x;                      // 0..255
    for (int p = 0; p < 128; ++p) {
        float pivv = A[p * 256 + p];
        __syncthreads();                            // all read pivot before write
        float prow = A[p * 256 + t] / pivv;
        A[p * 256 + t] = prow;
        __syncthreads();
        if (t < 128) mcol[t] = A[t * 256 + p];
        __syncthreads();
        for (int r = 0; r < 128; ++r)
            if (r != p) A[r * 256 + t] -= mcol[r] * prow;
        __syncthreads();
    }
}

// ---------------------------------------------------------------------------
// 5) 9x9 Gaussian-derivative cross-correlation (SAME, zero pad) on pred and
//    |pred|; 4 channels out, EXTGRADFAC=10 folded in. 16x16 tiles + halo 4.
// ---------------------------------------------------------------------------
__global__ void grad_kernel(const float* __restrict__ pred, float* __restrict__ gimg) {
    __shared__ float tP[24][24];
    __shared__ float tA[24][24];
    __shared__ float wY[81], wX[81];
    const int b  = blockIdx.z;
    const int ty = threadIdx.y, tx = threadIdx.x;
    const int lin = ty * 16 + tx;
    if (lin == 0) {
        float g[9], dg[9], s = 0.0f;
        for (int i = 0; i < 9; ++i) { float tt = (float)(i - 4); g[i] = expf(-tt * tt / 8.0f); s += g[i]; }
        for (int i = 0; i < 9; ++i) { g[i] /= s; dg[i] = -((float)(i - 4)) / 4.0f * g[i]; }
        for (int i = 0; i < 9; ++i)
            for (int j = 0; j < 9; ++j) { wY[i * 9 + j] = dg[i] * g[j]; wX[i * 9 + j] = g[i] * dg[j]; }
    }
    const int y0 = blockIdx.y * 16 - 4, x0 = blockIdx.x * 16 - 4;
    const float* pb = pred + (size_t)b * HWSZ;
    for (int idx = lin; idx < 24 * 24; idx += 256) {
        int iy = idx / 24, ix = idx % 24;
        int gy = y0 + iy, gx = x0 + ix;
        float v = (gy >= 0 && gy < HH && gx >= 0 && gx < WW) ? pb[gy * WW + gx] : 0.0f;
        tP[iy][ix] = v;
        tA[iy][ix] = fabsf(v);
    }
    __syncthreads();
    float a0 = 0.f, a1 = 0.f, a2 = 0.f, a3 = 0.f;
#pragma unroll
    for (int i = 0; i < 9; ++i)
#pragma unroll
        for (int j = 0; j < 9; ++j) {
            float p  = tP[ty + i][tx + j];
            float pa = tA[ty + i][tx + j];
            float wy = wY[i * 9 + j], wx = wX[i * 9 + j];
            a0 += wy * p;  a1 += wx * p;
            a2 += wy * pa; a3 += wx * pa;
        }
    const int oy = blockIdx.y * 16 + ty, ox = blockIdx.x * 16 + tx;
    size_t base = (size_t)b * 4 * HWSZ + (size_t)oy * WW + ox;
    gimg[base]             = 10.0f * a0;
    gimg[base + HWSZ]      = 10.0f * a1;
    gimg[base + 2 * HWSZ]  = 10.0f * a2;
    gimg[base + 3 * HWSZ]  = 10.0f * a3;
}

// ---------------------------------------------------------------------------
// 6) Snake evolution: one block per sample, 8 waves (256 threads).
//    A-tiles (wave's 16x128 slice of Minv/Mwinv) are loop-invariant across
//    the 50 steps -> preloaded into registers (64 VGPR each). Inner loop is
//    pure ds_load (B from LDS) + v_wmma: minimal latency on the serial scan.
// ---------------------------------------------------------------------------
__device__ __forceinline__ void matvec_wmma_reg(const v2f* __restrict__ aReg,
                                                const float* __restrict__ Ulds,
                                                float* __restrict__ Plds,
                                                int wave, int lane) {
    v8f acc = {0.f, 0.f, 0.f, 0.f, 0.f, 0.f, 0.f, 0.f};
    const int m     = lane & 15;
    const int khalf = (lane >> 4) * 2;
#pragma unroll
    for (int kb = 0; kb < 32; ++kb) {
        const int k = kb * 4 + khalf;
        v2f b;
        b.x = Ulds[k * 16 + m];
        b.y = Ulds[(k + 1) * 16 + m];
        acc = __builtin_amdgcn_wmma_f32_16x16x4_f32(false, aReg[kb], false, b,
                                                    (short)0, acc, false, false);
    }
    const int rbase = wave * 16 + ((lane >> 4) << 3);
#pragma unroll
    for (int v = 0; v < 8; ++v)
        Plds[(rbase + v) * 16 + m] = acc[v];
}

__device__ __forceinline__ float bilin(const float* __restrict__ img, float y, float x) {
    y = fminf(fmaxf(y, 0.0f), (float)HH - 1.001f);
    x = fminf(fmaxf(x, 0.0f), (float)WW - 1.001f);
    int y0 = (int)floorf(y), x0 = (int)floorf(x);
    float wy = y - (float)y0, wx = x - (float)x0;
    const float* p = img + y0 * WW + x0;
    float v00 = p[0], v01 = p[1], v10 = p[WW], v11 = p[WW + 1];
    return v00 * (1.f - wy) * (1.f - wx) + v01 * (1.f - wy) * wx
         + v10 * wy * (1.f - wx)         + v11 * wy * wx;
}

__global__ void __launch_bounds__(256, 1)
snake_kernel(const float* __restrict__ gimg,
             const float* __restrict__ aug,
             const float* __restrict__ nodes,
             float* __restrict__ nout) {
    __shared__ float U[128 * 16];
    __shared__ float P[128 * 16];
    const int b    = blockIdx.x;
    const int tid  = threadIdx.x;                   // 256 threads, 8 waves
    const int wave = tid >> 5, lane = tid & 31;
    const float* gy  = gimg + ((size_t)b * 4 + 0) * HWSZ;
    const float* gx  = gimg + ((size_t)b * 4 + 1) * HWSZ;
    const float* gwy = gimg + ((size_t)b * 4 + 2) * HWSZ;
    const float* gwx = gimg + ((size_t)b * 4 + 3) * HWSZ;
    const float* Minv  = aug + 128;                 // right half of AUG1 (stride 256)
    const float* Mwinv = aug + 32768 + 128;         // right half of AUG2

    // --- preload loop-invariant A-tiles into registers ---------------------
    v2f aM[32], aW[32];
    {
        const int m     = lane & 15;
        const int r     = wave * 16 + m;
        const int khalf = (lane >> 4) * 2;
#pragma unroll
        for (int kb = 0; kb < 32; ++kb) {
            const int k = kb * 4 + khalf;
            aM[kb].x = Minv [r * 256 + k];
            aM[kb].y = Minv [r * 256 + k + 1];
            aW[kb].x = Mwinv[r * 256 + k];
            aW[kb].y = Mwinv[r * 256 + k + 1];
        }
    }

    float py = 0.f, px = 0.f, wd = 1.0f;
    if (tid < 128) {
        py = nodes[(b * NN + tid) * 2 + 0];
        px = nodes[(b * NN + tid) * 2 + 1];
    }
    for (int s = 0; s < NSTEPS_; ++s) {
        if (tid < 128) {
            float fy = bilin(gy, py, px);
            float fx = bilin(gx, py, px);
            U[tid * 16 + 0] = py + 0.1f * fy;
            U[tid * 16 + 1] = px + 0.1f * fx;
        }
        __syncthreads();
        matvec_wmma_reg(aM, U, P, wave, lane);      // pts = M @ (pts + 0.1 f)
        __syncthreads();
        if (tid < 128) {
            py = P[tid * 16 + 0];
            px = P[tid * 16 + 1];
            float ay = bilin(gwy, py, px);
            float ax = bilin(gwx, py, px);
            U[tid * 16 + 0] = wd + 0.1f * sqrtf(ay * ay + ax * ax);
        }
        __syncthreads();
        matvec_wmma_reg(aW, U, P, wave, lane);      // wid = Mw @ (wid + 0.1 fw)
        __syncthreads();
        if (tid < 128) wd = P[tid * 16 + 0];
    }
    if (tid < 128) {
        nout[(b * NN + tid) * 3 + 0] = py;
        nout[(b * NN + tid) * 3 + 1] = px;
        nout[(b * NN + tid) * 3 + 2] = wd;
    }
}

// ---------------------------------------------------------------------------
// 7) Render min-distance map, squared error, per-block partial sums.
// ---------------------------------------------------------------------------
__global__ void render_loss_kernel(const float* __restrict__ pred,
                                   const float* __restrict__ nout,
                                   float* __restrict__ partials) {
    __shared__ float sy[128], sx[128], sw[128];
    __shared__ float red[256];
    const int b = blockIdx.y;
    const int tid = threadIdx.x;
    if (tid < 128) {
        sy[tid] = nout[(b * NN + tid) * 3 + 0];
        sx[tid] = nout[(b * NN + tid) * 3 + 1];
        sw[tid] = nout[(b * NN + tid) * 3 + 2];
    }
    __syncthreads();
    const int pix = blockIdx.x * 256 + tid;
    const float fy = (float)(pix / WW);
    const float fx = (float)(pix % WW);
    float dmin = 3.4e38f;
#pragma unroll 4
    for (int n = 0; n < NN; ++n) {
        float dy = fy - sy[n], dx = fx - sx[n];
        float d = sqrtf(dy * dy + dx * dx) - sw[n];
        dmin = fminf(dmin, d);
    }
    dmin = fminf(fmaxf(dmin, 0.0f), 15.0f);
    float e = pred[(size_t)b * HWSZ + pix] - dmin;
    red[tid] = e * e;
    __syncthreads();
    for (int s = 128; s > 0; s >>= 1) {
        if (tid < s) red[tid] += red[tid + s];
        __syncthreads();
    }
    if (tid == 0) partials[b * gridDim.x + blockIdx.x] = red[0];
}

// ---------------------------------------------------------------------------
// 8) Deterministic final reduction (fixed order, no float atomics).
// ---------------------------------------------------------------------------
__global__ void final_reduce_kernel(const float* __restrict__ partials,
                                    float* __restrict__ out) {
    __shared__ float red[256];
    const int tid = threadIdx.x;
    float s = 0.0f;
    for (int i = tid; i < 2304; i += 256) s += partials[i];
    red[tid] = s;
    __syncthreads();
    for (int k = 128; k > 0; k >>= 1) {
        if (tid < k) red[tid] += red[tid + k];
        __syncthreads();
    }
    if (tid == 0) out[0] = red[0] / (float)(BB * HWSZ);
}

// ---------------------------------------------------------------------------
extern "C" void kernel_launch(void* const* d_in, const int* in_sizes, int n_in,
                              void* d_out, int out_size, void* d_ws, size_t ws_size,
                              hipStream_t stream) {
    (void)in_sizes; (void)n_in; (void)out_size; (void)ws_size;
    const float* pred  = (const float*)d_in[0];   // B*1*H*W f32
    const float* nodes = (const float*)d_in[1];   // B*N*2  f32
    float* out = (float*)d_out;
    float* ws  = (float*)d_ws;

    float* Lm    = ws + OFF_L;
    float* LL    = ws + OFF_LL;
    float* aug   = ws + OFF_AUG;
    float* nout  = ws + OFF_NOUT;
    float* part  = ws + OFF_PART;
    float* gimg  = ws + OFF_GIMG;

    build_L_kernel    <<<64, 256, 0, stream>>>(Lm);
    ll_matmul_kernel  <<<64, 32, 0, stream>>>(Lm, LL);
    assemble_kernel   <<<256, 256, 0, stream>>>(Lm, LL, aug);
    gauss_jordan_kernel<<<2, 256, 0, stream>>>(aug);
    grad_kernel       <<<dim3(24, 24, BB), dim3(16, 16), 0, stream>>>(pred, gimg);
    snake_kernel      <<<BB, 256, 0, stream>>>(gimg, aug, nodes, nout);
    render_loss_kernel<<<dim3(HWSZ / 256, BB), 256, 0, stream>>>(pred, nout, part);
    final_reduce_kernel<<<1, 256, 0, stream>>>(part, out);
}